// CNTF_83683142795462
// MI455X (gfx1250) — compile-verified
//
#include <hip/hip_runtime.h>
#include <hip/hip_bf16.h>
#include <math.h>

#define T_DIM 512
#define RANK  32

typedef float v2f  __attribute__((ext_vector_type(2)));
typedef float v8f  __attribute__((ext_vector_type(8)));
typedef unsigned int u32x4 __attribute__((ext_vector_type(4)));
typedef int  i32x8 __attribute__((ext_vector_type(8)));
typedef int  i32x4 __attribute__((ext_vector_type(4)));

#if __has_builtin(__builtin_amdgcn_readlane)
#define CNTF_BCAST(v, t) __builtin_amdgcn_readlane((v), (t))
#else
#define CNTF_BCAST(v, t) __shfl((v), (t), 32)
#endif

// Scratch layout (floats): [0..31]=colsum(Ws) [32..63]=colsum(Ul) [64..95]=colsum(Um)
// [96]=unused [97]=sum_n vals*log(clip(A_sum))
__global__ void cntf_init(float* __restrict__ ws) {
    int t = threadIdx.x;
    if (t < 128) ws[t] = 0.0f;
}

// Column sums of X (nrows x 32) via f32 WMMA: D = ones(16x4) * B(4x16) + C.
// Every row of C holds the colsum partials; lanes 0-15 merge C[0] atomically.
__global__ void cntf_colsum(const float* __restrict__ X, int nrows,
                            float* __restrict__ out) {
    const int lane = threadIdx.x & 31;
    const int wv   = (blockIdx.x * blockDim.x + threadIdx.x) >> 5;
    const int nw   = (gridDim.x * blockDim.x) >> 5;
    const int colh = lane & 15;         // N within 16-wide half
    const int krow = (lane >> 4) << 1;  // lanes 0-15 -> K=0,1 ; lanes 16-31 -> K=2,3
#if __has_builtin(__builtin_amdgcn_wmma_f32_16x16x4_f32)
    v8f c0 = {0.f,0.f,0.f,0.f,0.f,0.f,0.f,0.f};
    v8f c1 = {0.f,0.f,0.f,0.f,0.f,0.f,0.f,0.f};
    v2f a; a.x = 1.0f; a.y = 1.0f;      // A = all-ones 16x4
    for (int rb = wv * 4; rb + 3 < nrows; rb += nw * 4) {
        const float* r0 = X + (size_t)(rb + krow) * RANK;
        v2f b0, b1;
        b0.x = r0[colh];        b0.y = r0[RANK + colh];        // cols 0-15, K/K+1
        b1.x = r0[16 + colh];   b1.y = r0[RANK + 16 + colh];   // cols 16-31
        c0 = __builtin_amdgcn_wmma_f32_16x16x4_f32(false, a, false, b0,
                                                   (short)0, c0, false, false);
        c1 = __builtin_amdgcn_wmma_f32_16x16x4_f32(false, a, false, b1,
                                                   (short)0, c1, false, false);
    }
    if (lane < 16) {
        atomicAdd(&out[colh],      c0[0]);   // C element (M=0 or 8, N=colh): colsum
        atomicAdd(&out[16 + colh], c1[0]);
    }
#else
    float s = 0.f;
    for (int r = wv; r < nrows; r += nw) s += X[(size_t)r * RANK + lane];
    atomicAdd(&out[lane], s);
#endif
}

// Butterfly combine for the multi-reduce: u covers nonzeros [t0, t0+m),
// v covers [t0+m, t0+2m); result covers [t0, t0+2m) with the invariant
// "lane L accumulates toward nonzero t0 + (L mod group)".
__device__ __forceinline__ float cntf_combine(float u, float v, int m, int lane) {
    float mine  = (lane & m) ? v : u;   // the vector my lane's target t lives in
    float other = (lane & m) ? u : v;   // the vector my xor-partner needs
    return mine + __shfl_xor(other, m, 32);
}

// Main gather kernel: one wave processes 32 nonzeros per outer iteration.
// Lane r owns rank r during gathers; per nonzero: 1 LDS read (Ws) + 2 coalesced
// 128B L2 gathers (scalar base + lane offset). Rank reduction across the batch
// uses a 31-combine butterfly multi-reduce, ending with lane L holding A_sum of
// nonzero base+L so all 32 logs run in parallel.
__global__ void __launch_bounds__(256)
cntf_main(const float* __restrict__ Ws, const float* __restrict__ Ul,
          const float* __restrict__ Um, const float* __restrict__ vals,
          const int* __restrict__ s0, const int* __restrict__ s1,
          const int* __restrict__ s2, long long nnz, float* __restrict__ acc) {
    __shared__ float wsl[T_DIM * RANK];   // 64 KB: entire Ws factor matrix

#if __has_builtin(__builtin_amdgcn_tensor_load_to_lds)
    if ((threadIdx.x >> 5) == 0) {        // wave 0 issues one TDM descriptor
        unsigned long long ga = (unsigned long long)(size_t)Ws;
        unsigned int lds_off  = (unsigned int)(size_t)&wsl[0]; // low 32b = LDS offset
        u32x4 g0;
        g0[0] = 1u;                                   // count=1, user descriptor
        g0[1] = lds_off;                              // lds_addr
        g0[2] = (unsigned int)ga;                     // global_addr[31:0]
        g0[3] = ((unsigned int)(ga >> 32) & 0x01FFFFFFu) | 0x80000000u; // [56:32]|type=2
        i32x8 g1;
        g1[0] = 0x00020000;        // wg_mask=0, data_size=4B
        g1[1] = 0x40000000;        // tensor_dim0 = 16384 (bits 63:48)
        g1[2] = 0x00010000;        // tensor_dim1 = 1
        g1[3] = 0x40000000;        // tile_dim0 = 16384 (bits 127:112)
        g1[4] = 0;                 // tile_dim1 = tile_dim2 = 0 (1-D tile)
        g1[5] = 16384;             // tensor_dim0_stride (low 32)
        g1[6] = 0;
        g1[7] = 0;
        i32x4 z4 = {0, 0, 0, 0};
        i32x8 z8 = {0, 0, 0, 0, 0, 0, 0, 0};
        __builtin_amdgcn_tensor_load_to_lds(g0, g1, z4, z4, z8, 0);
        __builtin_amdgcn_s_wait_tensorcnt(0);
    }
#else
    for (int i = threadIdx.x; i < T_DIM * RANK; i += blockDim.x) wsl[i] = Ws[i];
#endif
    __syncthreads();

    const int lane = threadIdx.x & 31;
    const long long gw = (long long)((blockIdx.x * blockDim.x + threadIdx.x) >> 5);
    const long long nw = (long long)((gridDim.x * blockDim.x) >> 5);
    float accl = 0.0f;

    for (long long base = gw * 32; base < nnz; base += nw * 32) {
        long long n = base + (long long)lane;
        bool ok = (n < nnz);
        long long nc = ok ? n : (nnz - 1);
        int   iv = s0[nc];                     // coalesced index/val loads
        int   jv = s1[nc];
        int   kv = s2[nc];
        float vv = ok ? vals[nc] : 0.0f;

        float lvl[5];                          // binary-counter merge stack
        float fin = 0.0f;
#pragma unroll
        for (int t = 0; t < 32; ++t) {
            int i = CNTF_BCAST(iv, t);         // v_readlane -> wave-uniform SGPR
            int j = CNTF_BCAST(jv, t);
            int k = CNTF_BCAST(kv, t);
            float v = wsl[i * RANK + lane]     // conflict-free ds_load
                    * Ul[(size_t)j * RANK + lane]   // one 128B line per wave
                    * Um[(size_t)k * RANK + lane];
            // merge like a binary increment: level b absorbs groups of 2^b
            bool stored = false;
#pragma unroll
            for (int b = 0; b < 5; ++b) {
                if (!stored) {
                    if (t & (1 << b)) {
                        v = cntf_combine(lvl[b], v, 1 << b, lane);
                    } else {
                        lvl[b] = v;
                        stored = true;
                    }
                }
            }
            if (t == 31) fin = v;              // lane L now holds A_sum[base+L]
        }
        accl += vv * logf(fmaxf(fin, 1e-10f)); // 32 logs fully parallel
    }
    accl += __shfl_xor(accl, 16, 32);
    accl += __shfl_xor(accl,  8, 32);
    accl += __shfl_xor(accl,  4, 32);
    accl += __shfl_xor(accl,  2, 32);
    accl += __shfl_xor(accl,  1, 32);
    if (lane == 0) atomicAdd(acc, accl);
}

__global__ void cntf_finalize(const float* __restrict__ wsbuf,
                              float* __restrict__ out) {
    const int lane = threadIdx.x & 31;
    float s = wsbuf[lane] * wsbuf[32 + lane] * wsbuf[64 + lane];
    s += __shfl_xor(s, 16, 32);
    s += __shfl_xor(s,  8, 32);
    s += __shfl_xor(s,  4, 32);
    s += __shfl_xor(s,  2, 32);
    s += __shfl_xor(s,  1, 32);
    if (threadIdx.x == 0) {
        float ll = (wsbuf[97] - s) / (float)T_DIM;
        out[0] = -ll;
    }
}

extern "C" void kernel_launch(void* const* d_in, const int* in_sizes, int n_in,
                              void* d_out, int out_size, void* d_ws, size_t ws_size,
                              hipStream_t stream) {
    const float* Ws   = (const float*)d_in[0];
    const float* Ul   = (const float*)d_in[1];
    const float* Um   = (const float*)d_in[2];
    const float* vals = (const float*)d_in[3];
    const int*   s0   = (const int*)d_in[4];
    const int*   s1   = (const int*)d_in[5];
    const int*   s2   = (const int*)d_in[6];
    long long nnz = (long long)in_sizes[3];
    int nt = in_sizes[0] / RANK;
    int nl = in_sizes[1] / RANK;
    int nm = in_sizes[2] / RANK;
    float* wsf = (float*)d_ws;
    float* out = (float*)d_out;

    cntf_init<<<1, 128, 0, stream>>>(wsf);
    cntf_colsum<<<8,  256, 0, stream>>>(Ws, nt, wsf + 0);
    cntf_colsum<<<64, 256, 0, stream>>>(Ul, nl, wsf + 32);
    cntf_colsum<<<32, 256, 0, stream>>>(Um, nm, wsf + 64);
    cntf_main<<<4096, 256, 0, stream>>>(Ws, Ul, Um, vals, s0, s1, s2, nnz, wsf + 97);
    cntf_finalize<<<1, 32, 0, stream>>>(wsf, out);
}